// Loss_86414741995528
// MI455X (gfx1250) — compile-verified
//
#include <hip/hip_runtime.h>
#include <hip/hip_bf16.h>

// ---------------------------------------------------------------------------
// Triplet loss w/ batch-hard mining on MI455X (gfx1250).
//   G = X * X^T via V_WMMA_F32_16X16X4_F32 (native fp32 matrix pipe)
//   dist(i,j) = sqrt(max(sq_i + sq_j - 2*G_ij, 1e-12))
//   dist_ap[i] = max over same-label j ; dist_an[i] = min over diff-label j
//   loss = mean(relu(dist_ap - dist_an + 0.3))
// ---------------------------------------------------------------------------

#define NROWS 8192
#define FEAT  128
#define TILE  128          // macro tile per block (128x128 outputs)
#define KCH   32           // K staged through LDS per chunk
#define LPAD  34           // LDS row stride (even -> b64 aligned; 34 mod 64 avoids conflicts)
#define MARGIN 0.3f

typedef float v2f __attribute__((ext_vector_type(2)));
typedef float v8f __attribute__((ext_vector_type(8)));

// ---------------------------------------------------------------------------
// Kernel A: row squared-norms + reduction-buffer init (re-run every launch so
// the workspace never carries state across calls).
// ---------------------------------------------------------------------------
__global__ __launch_bounds__(256) void trip_init_kernel(
    const float* __restrict__ X, float* __restrict__ sq,
    unsigned* __restrict__ ap_bits, unsigned* __restrict__ an_bits)
{
    int r = blockIdx.x * blockDim.x + threadIdx.x;     // one row per thread
    if (r >= NROWS) return;
    const float4* row = (const float4*)(X + (size_t)r * FEAT);
    float acc = 0.f;
#pragma unroll
    for (int i = 0; i < FEAT / 4; ++i) {
        float4 v = row[i];
        acc += v.x * v.x + v.y * v.y + v.z * v.z + v.w * v.w;
    }
    sq[r]      = acc;
    ap_bits[r] = 0u;           // +0.0f : identity for max of non-negative dists
    an_bits[r] = 0x7F800000u;  // +inf  : identity for min
}

// ---------------------------------------------------------------------------
// Kernel B: 128x128 macro-tile. 8 wave32s; wave w owns rows [w*16, w*16+16)
// of the I-tile against all 128 J columns (8 sub-tiles of 16x16).
// Gram accumulated with V_WMMA_F32_16X16X4_F32, K staged via LDS.
// ---------------------------------------------------------------------------
__global__ __launch_bounds__(256) void trip_tile_kernel(
    const float* __restrict__ X, const int* __restrict__ tgt,
    const float* __restrict__ sq,
    unsigned* __restrict__ ap_bits, unsigned* __restrict__ an_bits)
{
    __shared__ float lA[TILE][LPAD];   // I-tile rows, current K chunk
    __shared__ float lB[TILE][LPAD];   // J-tile rows, current K chunk
    __shared__ float sqI[TILE], sqJ[TILE];
    __shared__ int   tI[TILE],  tJ[TILE];

    const int i0   = blockIdx.y * TILE;
    const int j0   = blockIdx.x * TILE;
    const int tid  = threadIdx.x;
    const int lane = tid & 31;
    const int wave = tid >> 5;       // 0..7
    const int half = lane >> 4;      // 0 | 1  (K-pair select per ISA A/B layout)
    const int ln   = lane & 15;      // row-in-subtile / column-in-subtile

    // Stage per-row scalars (norms + labels) once.
    if (tid < TILE) {
        sqI[tid] = sq[i0 + tid];
        tI[tid]  = tgt[i0 + tid];
    } else {
        int t = tid - TILE;
        sqJ[t] = sq[j0 + t];
        tJ[t]  = tgt[j0 + t];
    }

    v8f acc[8];
#pragma unroll
    for (int jt = 0; jt < 8; ++jt) {
        v8f z = {0.f, 0.f, 0.f, 0.f, 0.f, 0.f, 0.f, 0.f};
        acc[jt] = z;
    }

    for (int k0 = 0; k0 < FEAT; k0 += KCH) {
        __syncthreads();   // previous chunk's readers done before overwrite
        // Stage 128 rows x 32 floats for each tile: 1024 float4 / tile,
        // 4 float4 per thread per tile.
#pragma unroll
        for (int it = 0; it < 4; ++it) {
            int idx = tid + it * 256;          // 0..1023
            int row = idx >> 3;                // 0..127
            int c4  = (idx & 7) << 2;          // 0,4,...,28
            float4 va = *(const float4*)(X + (size_t)(i0 + row) * FEAT + k0 + c4);
            float4 vb = *(const float4*)(X + (size_t)(j0 + row) * FEAT + k0 + c4);
            lA[row][c4 + 0] = va.x; lA[row][c4 + 1] = va.y;
            lA[row][c4 + 2] = va.z; lA[row][c4 + 3] = va.w;
            lB[row][c4 + 0] = vb.x; lB[row][c4 + 1] = vb.y;
            lB[row][c4 + 2] = vb.z; lB[row][c4 + 3] = vb.w;
            if (k0 + KCH < FEAT) {  // pull next chunk toward L2/L0 early
                __builtin_prefetch(X + (size_t)(i0 + row) * FEAT + k0 + KCH + c4, 0, 0);
                __builtin_prefetch(X + (size_t)(j0 + row) * FEAT + k0 + KCH + c4, 0, 0);
            }
        }
        __syncthreads();   // stores visible before fragment reads

#pragma unroll
        for (int k = 0; k < KCH; k += 4) {
            const int kk = k + (half << 1);    // K pair for this lane half
            // A fragment 16x4: lane (half,ln) holds A[ln][kk], A[ln][kk+1]
            v2f afrag;
            afrag[0] = lA[wave * 16 + ln][kk];
            afrag[1] = lA[wave * 16 + ln][kk + 1];
#pragma unroll
            for (int jt = 0; jt < 8; ++jt) {
                // B fragment 4x16: lane (half,ln) holds B[kk][ln], B[kk+1][ln]
                // with B = X_Jtile^T  =>  read rows of the J tile.
                v2f bfrag;
                bfrag[0] = lB[jt * 16 + ln][kk];
                bfrag[1] = lB[jt * 16 + ln][kk + 1];
                acc[jt] = __builtin_amdgcn_wmma_f32_16x16x4_f32(
                    /*neg_a=*/false, afrag, /*neg_b=*/false, bfrag,
                    /*c_mod=*/(short)0, acc[jt],
                    /*reuse_a=*/false, /*reuse_b=*/false);
            }
        }
    }

    // Epilogue: distances, mask, per-row hard mining.
    // C/D layout: VGPR r <-> row (8*half + r), lane&15 <-> column.
    float apv[8], anv[8];
#pragma unroll
    for (int r = 0; r < 8; ++r) { apv[r] = 0.f; anv[r] = __builtin_inff(); }

#pragma unroll
    for (int jt = 0; jt < 8; ++jt) {
        const float sqj = sqJ[jt * 16 + ln];
        const int   tj  = tJ[jt * 16 + ln];
#pragma unroll
        for (int r = 0; r < 8; ++r) {
            const int mi  = half * 8 + r;               // row within wave's 16
            const float g = acc[jt][r];
            float d2   = sqI[wave * 16 + mi] + sqj - 2.f * g;
            float dist = __builtin_sqrtf(fmaxf(d2, 1e-12f));
            if (tI[wave * 16 + mi] == tj) apv[r] = fmaxf(apv[r], dist);
            else                          anv[r] = fminf(anv[r], dist);
        }
    }

    // Reduce across the 16 lanes sharing each row (xor masks <16 stay within
    // each half-wave), then one atomic per row per block on the bit pattern
    // (valid: dists are non-negative -> unsigned order == float order).
#pragma unroll
    for (int r = 0; r < 8; ++r) {
        float a = apv[r], b = anv[r];
#pragma unroll
        for (int off = 1; off < 16; off <<= 1) {
            a = fmaxf(a, __shfl_xor(a, off, 32));
            b = fminf(b, __shfl_xor(b, off, 32));
        }
        if (ln == 0) {
            const int row = i0 + wave * 16 + half * 8 + r;
            atomicMax(&ap_bits[row], __float_as_uint(a));
            atomicMin(&an_bits[row], __float_as_uint(b));
        }
    }
}

// ---------------------------------------------------------------------------
// Kernel C: loss = mean(relu(dist_ap - dist_an + margin)). Single block,
// deterministic tree reduction.
// ---------------------------------------------------------------------------
__global__ __launch_bounds__(256) void trip_loss_kernel(
    const unsigned* __restrict__ ap_bits, const unsigned* __restrict__ an_bits,
    float* __restrict__ out)
{
    __shared__ float red[256];
    float acc = 0.f;
    for (int r = threadIdx.x; r < NROWS; r += 256) {
        float dap = __uint_as_float(ap_bits[r]);
        float dan = __uint_as_float(an_bits[r]);
        float v   = dap - dan + MARGIN;
        acc += (v > 0.f) ? v : 0.f;
    }
    red[threadIdx.x] = acc;
    __syncthreads();
#pragma unroll
    for (int s = 128; s > 0; s >>= 1) {
        if (threadIdx.x < (unsigned)s) red[threadIdx.x] += red[threadIdx.x + s];
        __syncthreads();
    }
    if (threadIdx.x == 0) out[0] = red[0] * (1.f / (float)NROWS);
}

// ---------------------------------------------------------------------------
extern "C" void kernel_launch(void* const* d_in, const int* in_sizes, int n_in,
                              void* d_out, int out_size, void* d_ws, size_t ws_size,
                              hipStream_t stream)
{
    (void)in_sizes; (void)n_in; (void)out_size; (void)ws_size;

    const float* X   = (const float*)d_in[0];   // [8192,128] fp32
    const int*   tgt = (const int*)d_in[1];     // [8192] labels
    float*       out = (float*)d_out;           // scalar

    // Workspace: ap_bits[N] | an_bits[N] | sq[N]  (96 KB)
    unsigned* ap = (unsigned*)d_ws;
    unsigned* an = ap + NROWS;
    float*    sq = (float*)(an + NROWS);

    trip_init_kernel<<<NROWS / 256, 256, 0, stream>>>(X, sq, ap, an);

    dim3 grid(NROWS / TILE, NROWS / TILE);      // 64 x 64 macro-tiles
    trip_tile_kernel<<<grid, 256, 0, stream>>>(X, tgt, sq, ap, an);

    trip_loss_kernel<<<1, 256, 0, stream>>>(ap, an, out);
}